// QCNN_58025008169534
// MI455X (gfx1250) — compile-verified
//
#include <hip/hip_runtime.h>

// ---------------------------------------------------------------------------
// QCNN forward for MI455X (gfx1250, wave32).
//   Kernel 1: fold the 4 shared variational layers (RZ*RY*RX per qubit + CZs)
//             into one 16x16 complex unitary V (stored in d_ws).
//   Kernel 2: per circuit, build the rank-1 encoded state (trig on VALU,
//             split across lane halves + cross-lane exchange), then batched
//             complex 16x16 matvec on the tensor cores via
//             V_WMMA_F32_16X16X4_F32 (16 circuits per wave, 16 wmma/tile).
// ---------------------------------------------------------------------------

typedef __attribute__((ext_vector_type(2))) float v2f;
typedef __attribute__((ext_vector_type(8))) float v8f;

#define IMG_B   16
#define IMG_H   128
#define IMG_W   128
#define P_POS   (127 * 127)          // patches per image
#define N_TILES 16129                // (16*127*127)/16 circuits per 16-wide tile
#define WAVES   8                    // waves per block

struct cplxf { float re, im; };
__device__ inline cplxf cmul(cplxf a, cplxf b) {
    return { a.re * b.re - a.im * b.im, a.re * b.im + a.im * b.re };
}
__device__ inline cplxf cadd(cplxf a, cplxf b) { return { a.re + b.re, a.im + b.im }; }

__device__ inline void mm2(const cplxf A[2][2], const cplxf B[2][2], cplxf C[2][2]) {
    for (int i = 0; i < 2; ++i)
        for (int j = 0; j < 2; ++j)
            C[i][j] = cadd(cmul(A[i][0], B[0][j]), cmul(A[i][1], B[1][j]));
}

// ------------------------- Kernel 1: variational unitary -------------------
// Thread c owns column c of V (V starts as identity; V <- G*V per gate).
// State index k: qubit q lives in bit (3-q).
__global__ void qcnn_unitary_kernel(const float* __restrict__ ang,
                                    float* __restrict__ Vr,
                                    float* __restrict__ Vi)
{
    const int c = threadIdx.x;
    if (c >= 16) return;

    cplxf v[16];
    for (int k = 0; k < 16; ++k) v[k] = { (k == c) ? 1.0f : 0.0f, 0.0f };

    for (int il = 0; il < 4; ++il) {
        for (int q = 0; q < 4; ++q) {
            const float a = ang[(il * 4 + q) * 3 + 0];
            const float b = ang[(il * 4 + q) * 3 + 1];
            const float g = ang[(il * 4 + q) * 3 + 2];
            const float ca = __cosf(0.5f * a), sa = __sinf(0.5f * a);
            const float cb = __cosf(0.5f * b), sb = __sinf(0.5f * b);
            const float cg = __cosf(0.5f * g), sg = __sinf(0.5f * g);

            const cplxf X[2][2] = { { { ca, 0.f }, { 0.f, -sa } },
                                    { { 0.f, -sa }, { ca, 0.f } } };   // RX
            const cplxf Y[2][2] = { { { cb, 0.f }, { -sb, 0.f } },
                                    { { sb, 0.f }, { cb, 0.f } } };    // RY
            const cplxf Z[2][2] = { { { cg, -sg }, { 0.f, 0.f } },
                                    { { 0.f, 0.f }, { cg, sg } } };    // RZ
            cplxf M[2][2], U[2][2];
            mm2(Y, X, M);      // RY*RX
            mm2(Z, M, U);      // RZ*RY*RX

            const int m = 1 << (3 - q);
            for (int i = 0; i < 16; ++i) {
                if (i & m) continue;
                const int j = i | m;
                const cplxf vi = v[i], vj = v[j];
                v[i] = cadd(cmul(U[0][0], vi), cmul(U[0][1], vj));
                v[j] = cadd(cmul(U[1][0], vi), cmul(U[1][1], vj));
            }
        }
        // CZ(0,1): bits 3&2 ; CZ(2,3): bits 1&0 ; then CZ(1,2): bits 2&1
        for (int k = 0; k < 16; ++k) {
            float s = 1.0f;
            if ((k & 8) && (k & 4)) s = -s;
            if ((k & 2) && (k & 1)) s = -s;
            if ((k & 4) && (k & 2)) s = -s;
            v[k].re *= s; v[k].im *= s;
        }
    }
    for (int k = 0; k < 16; ++k) {      // row-major V[M][K]
        Vr[k * 16 + c] = v[k].re;
        Vi[k * 16 + c] = v[k].im;
    }
}

// ------------------------- Kernel 2: main WMMA kernel ----------------------
__global__ __launch_bounds__(32 * WAVES)
void qcnn_wmma_kernel(const float* __restrict__ img,
                      const float* __restrict__ Vr,
                      const float* __restrict__ Vi,
                      float* __restrict__ out)
{
    __shared__ float Sr[WAVES][16 * 17];
    __shared__ float Si[WAVES][16 * 17];

    const int lane = threadIdx.x & 31;
    const int wave = threadIdx.x >> 5;
    const int c    = lane & 15;          // circuit within tile / matrix column
    const int h    = lane >> 4;          // lane half

    int tile = blockIdx.x * WAVES + wave;
    const bool valid = (tile < N_TILES);
    if (!valid) tile = 0;                // dummy work, keep EXEC uniform

    const int n  = tile * 16 + c;        // circuit index
    const int b  = n / P_POS;
    const int p  = n - b * P_POS;
    const int pi = p / 127;
    const int pj = p - pi * 127;

    // ---- encoding: lane half h computes qubits 2h and 2h+1 only ----
    // a_q = first column of Rot(pi*x0, pi*x1, pi*x2)
    cplxf aq[2][2];                      // [local qubit][bit]
    #pragma unroll
    for (int ql = 0; ql < 2; ++ql) {
        const int di = h, dj = ql;       // q = 2h + ql -> di = q>>1 = h, dj = q&1
        const float* px = img + (((b * IMG_H + pi + di) * IMG_W) + (pj + dj)) * 3;
        const float x0 = px[0], x1 = px[1], x2 = px[2];
        const float PIH = 1.57079632679489662f;      // pi/2
        const float t  = PIH * x1;                   // theta/2
        const float pp = PIH * (x0 + x2);            // (phi+omega)/2
        const float mm = PIH * (x0 - x2);            // (phi-omega)/2
        const float ct = __cosf(t), st = __sinf(t);
        aq[ql][0] = { ct * __cosf(pp), -ct * __sinf(pp) };
        aq[ql][1] = { st * __cosf(mm), -st * __sinf(mm) };
    }
    // pairwise products of this half's two qubits: g[i*2+j] = a_{2h}[i]*a_{2h+1}[j]
    cplxf g[4];
    #pragma unroll
    for (int i = 0; i < 2; ++i)
        #pragma unroll
        for (int j = 0; j < 2; ++j)
            g[i * 2 + j] = cmul(aq[0][i], aq[1][j]);

    // exchange with the partner half-lane (same circuit, other h)
    cplxf go[4];
    #pragma unroll
    for (int i = 0; i < 4; ++i) {
        go[i].re = __shfl_xor(g[i].re, 16, 32);
        go[i].im = __shfl_xor(g[i].im, 16, 32);
    }
    // top factor = qubits(0,1) products, bottom factor = qubits(2,3) products
    cplxf gt[4], gb[4];
    #pragma unroll
    for (int i = 0; i < 4; ++i) {
        gt[i] = h ? go[i] : g[i];
        gb[i] = h ? g[i]  : go[i];
    }

    // state_enc[k] = gt[k>>2] * gb[k&3]; this lane fills rows k = 8h..8h+7
    #pragma unroll
    for (int kk = 0; kk < 8; ++kk) {
        const int k = 8 * h + kk;
        const cplxf s = cmul(gt[k >> 2], gb[k & 3]);
        Sr[wave][k * 17 + c] = s.re;
        Si[wave][k * 17 + c] = s.im;
    }
    __syncthreads();

    // ---- Y = V * S (complex) via 16x f32 WMMA: K in 4 chunks of 4 ----
    v8f accR = { 0, 0, 0, 0, 0, 0, 0, 0 };
    v8f accI = { 0, 0, 0, 0, 0, 0, 0, 0 };
    #pragma unroll
    for (int kb = 0; kb < 16; kb += 4) {
        const int ka = kb + 2 * h;
        const v2f ar  = { Vr[c * 16 + ka], Vr[c * 16 + ka + 1] };
        const v2f ai  = { Vi[c * 16 + ka], Vi[c * 16 + ka + 1] };
        const v2f ain = { -ai.x, -ai.y };                    // f32 WMMA: no A-neg modifier
        const v2f br  = { Sr[wave][ka * 17 + c], Sr[wave][(ka + 1) * 17 + c] };
        const v2f bi  = { Si[wave][ka * 17 + c], Si[wave][(ka + 1) * 17 + c] };
        accR = __builtin_amdgcn_wmma_f32_16x16x4_f32(false, ar,  false, br, (short)0, accR, false, false);
        accR = __builtin_amdgcn_wmma_f32_16x16x4_f32(false, ain, false, bi, (short)0, accR, false, false);
        accI = __builtin_amdgcn_wmma_f32_16x16x4_f32(false, ar,  false, bi, (short)0, accI, false, false);
        accI = __builtin_amdgcn_wmma_f32_16x16x4_f32(false, ai,  false, br, (short)0, accI, false, false);
    }

    // ---- probs -> <Z_q>; lane holds rows k = r + 8h of circuit c ----
    float ev[4] = { 0.f, 0.f, 0.f, 0.f };
    #pragma unroll
    for (int r = 0; r < 8; ++r) {
        const int k = r + 8 * h;
        const float pr = accR[r] * accR[r] + accI[r] * accI[r];
        ev[0] += (k & 8) ? -pr : pr;   // qubit 0 -> bit 3
        ev[1] += (k & 4) ? -pr : pr;   // qubit 1 -> bit 2
        ev[2] += (k & 2) ? -pr : pr;   // qubit 2 -> bit 1
        ev[3] += (k & 1) ? -pr : pr;   // qubit 3 -> bit 0
    }
    #pragma unroll
    for (int q = 0; q < 4; ++q) ev[q] += __shfl_xor(ev[q], 16, 32);

    if (valid && h == 0) {
        float4 o = { ev[0], ev[1], ev[2], ev[3] };
        reinterpret_cast<float4*>(out)[n] = o;
    }
}

// ------------------------- launcher ----------------------------------------
extern "C" void kernel_launch(void* const* d_in, const int* in_sizes, int n_in,
                              void* d_out, int out_size, void* d_ws, size_t ws_size,
                              hipStream_t stream)
{
    const float* img = (const float*)d_in[0];   // [16,128,128,3] f32
    const float* ang = (const float*)d_in[1];   // [4,4,3] f32
    float* out = (float*)d_out;                 // [16,16129,4] f32
    float* Vr  = (float*)d_ws;                  // 256 f32
    float* Vi  = Vr + 256;                      // 256 f32

    qcnn_unitary_kernel<<<1, 16, 0, stream>>>(ang, Vr, Vi);

    const int blocks = (N_TILES + WAVES - 1) / WAVES;   // 2017
    qcnn_wmma_kernel<<<blocks, 32 * WAVES, 0, stream>>>(img, Vr, Vi, out);
}